// AllAtomFAPE_43370579755546
// MI455X (gfx1250) — compile-verified
//
#include <hip/hip_runtime.h>
#include <hip/hip_bf16.h>

// ---------------------------------------------------------------------------
// AllAtomFAPE on gfx1250 (MI455X, wave32, WMMA).
//
// Delta(f,a) = M[f] @ u[a] with M[f] (3x8) = [Rp^T | -Rt^T | tinv_p - tinv_t | 0],
// u[a] (8)   = [pp.xyz, pt.xyz, 1, 0].
// Frames padded to 4 rows -> one 16x16 D tile = 4 frames x 16 atoms, produced
// by two V_WMMA_F32_16X16X4_F32 (K=8 split into two K=4 steps).
//
// A-operand layout (ISA 7.12.2, 32-bit A 16x4): lanes 0-15 hold row=lane with
// {K0,K1} in {v0,v1}; lanes 16-31 hold row=lane-16 with {K2,K3}.
// B-operand layout analogous (rows K0/K2 across lane halves in VGPR0, K1/K3
// in VGPR1). C/D layout: VGPR r = rows r (lanes 0-15) and r+8 (lanes 16-31).
// ---------------------------------------------------------------------------

typedef __attribute__((ext_vector_type(2))) float v2f;
typedef __attribute__((ext_vector_type(8))) float v8f;

static constexpr int NRES = 384;
static constexpr int NFPR = 8;            // frames per residue
static constexpr int NAPR = 14;           // atoms per residue
static constexpr int NFR  = NRES * NFPR;  // 3072 frames
static constexpr int NAT  = NRES * NAPR;  // 5376 atoms
static constexpr int FG   = NFR / 4;      // 768 frame groups (4 frames/tile)
static constexpr int AT   = NAT / 16;     // 336 atom tiles
static constexpr float FAPE_EPS   = 1e-4f;
static constexpr float FAPE_CLAMP = 10.0f;
static constexpr float FAPE_Z     = 10.0f;

// ws layout (floats): [0]=total, [1]=atom_count, [2]=frame_count,
// [16 .. 16+NAT) = optional combined atom-mask plane.
static constexpr int WS_PLANE_OFF = 16;

__global__ void fape_init(float* ws) {
  if (threadIdx.x == 0) { ws[0] = 0.f; ws[1] = 0.f; ws[2] = 0.f; }
}

// ws[1] = sum(atom_mask*seq_mask), ws[2] = sum(frame_mask) = 8*sum(seq_mask);
// optionally materializes the combined atom-mask plane (hoists the /14 and
// one extra load out of the 336-iteration WMMA loop).
__global__ void fape_counts(const float* __restrict__ am,
                            const float* __restrict__ sm,
                            float* __restrict__ ws, int write_plane) {
  int i = blockIdx.x * blockDim.x + threadIdx.x;
  if (i < NAT) {
    float amk = am[i] * sm[i / NAPR];
    atomicAdd(&ws[1], amk);
    if (write_plane) ws[WS_PLANE_OFF + i] = amk;
  }
  if (i < NRES) atomicAdd(&ws[2], (float)NFPR * sm[i]);
}

template <bool USE_PLANE>
__global__ __launch_bounds__(256) void fape_main(
    const float* __restrict__ Rp, const float* __restrict__ tp,
    const float* __restrict__ pp, const float* __restrict__ am,
    const float* __restrict__ Rt, const float* __restrict__ tt,
    const float* __restrict__ pt, const float* __restrict__ sm,
    float* __restrict__ ws) {
  const int lane = threadIdx.x & 31;
  const int wave = threadIdx.x >> 5;
  const int g    = blockIdx.x * 8 + wave;  // frame group, 0..767 (exact grid)
  const int half = lane >> 4;              // 0: K0..1 / rows 0..7 ; 1: K2..3 / rows 8..15
  const int row  = lane & 15;              // tile row (A) / column n (B,C)
  const int comp = row & 3;                // component within frame; 3 == zero pad row
  const int f    = g * 4 + (row >> 2);     // this lane's A-matrix frame
  const float* __restrict__ amk_plane = ws + WS_PLANE_OFF;

  // ---- Build A operands (M[f], rows padded with zeros at comp==3) ----------
  auto Mval = [&](int j) -> float {        // M[f][comp][j]
    if (comp == 3) return 0.f;
    if (j < 3) return Rp[f * 9 + j * 3 + comp];          // Rp^T
    if (j < 6) return -Rt[f * 9 + (j - 3) * 3 + comp];   // -Rt^T
    if (j == 6) {                                        // tinv_p - tinv_t
      float s = 0.f;
      for (int jj = 0; jj < 3; ++jj)
        s += Rt[f * 9 + jj * 3 + comp] * tt[f * 3 + jj]
           - Rp[f * 9 + jj * 3 + comp] * tp[f * 3 + jj];
      return s;
    }
    return 0.f;                                          // column 7 (u pad)
  };
  const int c0 = half * 2;                 // this lane's first K column
  v2f A0 = {Mval(c0 + 0), Mval(c0 + 1)};   // K = 0..3 chunk
  v2f A1 = {Mval(c0 + 4), Mval(c0 + 5)};   // K = 4..7 chunk

  // Frames covered by this lane's C registers: rows {0..7} or {8..15}.
  const int fA = g * 4 + half * 2;
  const int fB = fA + 1;
  const float wA = sm[fA >> 3];            // frame_mask = seq_mask[frame/8]
  const float wB = sm[fB >> 3];

  // ---- Sweep all atom tiles ------------------------------------------------
  float accA = 0.f, accB = 0.f;
  for (int t = 0; t < AT; ++t) {
    const int n = t * 16 + row;            // atom column for this lane
    const float u0 = pp[n * 3 + 0], u1 = pp[n * 3 + 1], u2 = pp[n * 3 + 2];
    const float u3 = pt[n * 3 + 0], u4 = pt[n * 3 + 1], u5 = pt[n * 3 + 2];
    v2f B0, B1;                            // B[k][n]: rows per lane half
    B0.x = half ? u2   : u0;   B0.y = half ? u3   : u1;   // K {0,1} / {2,3}
    B1.x = half ? 1.0f : u4;   B1.y = half ? 0.0f : u5;   // K {4,5} / {6,7}

    v8f c = {};
    c = __builtin_amdgcn_wmma_f32_16x16x4_f32(false, A0, false, B0,
                                              (short)0, c, false, false);
    c = __builtin_amdgcn_wmma_f32_16x16x4_f32(false, A1, false, B1,
                                              (short)0, c, false, false);

    const float amk = USE_PLANE ? amk_plane[n] : am[n] * sm[n / NAPR];
    // Raw v_sqrt_f32 (~1 ulp) instead of the IEEE-correct libm expansion;
    // values are O(1..100) loss distances, so the Newton fixup chain is waste.
    const float dA = __builtin_amdgcn_sqrtf(c[0] * c[0] + c[1] * c[1] +
                                            c[2] * c[2] + FAPE_EPS);
    const float dB = __builtin_amdgcn_sqrtf(c[4] * c[4] + c[5] * c[5] +
                                            c[6] * c[6] + FAPE_EPS);
    accA += fminf(dA, FAPE_CLAMP) * amk;
    accB += fminf(dB, FAPE_CLAMP) * amk;
  }

  // ---- Frame-mask weight, wave reduction, one atomic per wave --------------
  float val = accA * wA + accB * wB;
  #pragma unroll
  for (int m = 16; m >= 1; m >>= 1) val += __shfl_xor(val, m, 32);
  if (lane == 0) atomicAdd(&ws[0], val);
}

__global__ void fape_final(const float* __restrict__ ws, float* __restrict__ out) {
  if (threadIdx.x == 0 && blockIdx.x == 0) {
    const float ac = fmaxf(ws[1], 1.0f);
    const float fc = fmaxf(ws[2], 1.0f);
    out[0] = ws[0] / (ac * fc * FAPE_Z);
  }
}

extern "C" void kernel_launch(void* const* d_in, const int* in_sizes, int n_in,
                              void* d_out, int out_size, void* d_ws, size_t ws_size,
                              hipStream_t stream) {
  (void)in_sizes; (void)n_in; (void)out_size;
  const float* Rp = (const float*)d_in[0];
  const float* tp = (const float*)d_in[1];
  const float* pp = (const float*)d_in[2];
  const float* am = (const float*)d_in[3];
  const float* Rt = (const float*)d_in[4];
  const float* tt = (const float*)d_in[5];
  const float* pt = (const float*)d_in[6];
  const float* sm = (const float*)d_in[7];
  float* ws  = (float*)d_ws;
  float* out = (float*)d_out;

  const bool use_plane = ws_size >= (size_t)(WS_PLANE_OFF + NAT) * sizeof(float);

  fape_init<<<1, 32, 0, stream>>>(ws);
  fape_counts<<<(NAT + 255) / 256, 256, 0, stream>>>(am, sm, ws,
                                                     use_plane ? 1 : 0);
  if (use_plane)
    fape_main<true><<<FG / 8, 256, 0, stream>>>(Rp, tp, pp, am, Rt, tt, pt, sm, ws);
  else
    fape_main<false><<<FG / 8, 256, 0, stream>>>(Rp, tp, pp, am, Rt, tt, pt, sm, ws);
  fape_final<<<1, 32, 0, stream>>>(ws, out);
}